// MultiHeadWKV_50379966382717
// MI455X (gfx1250) — compile-verified
//
#include <hip/hip_runtime.h>

typedef __attribute__((ext_vector_type(2))) float v2f;
typedef __attribute__((ext_vector_type(8))) float v8f;

#define D_   1024
#define H_   16
#define dh   64
#define T_   4096
#define CH   16
#define SEG  16            // segments per head (parallel over the chip)
#define TSEG (T_/SEG)      // 256 timesteps per segment
#define NCHS (TSEG/CH)     // 16 chunks per segment
#define ST   68            // padded stride, 64-wide LDS rows (conflict-free)
#define ST16 20            // padded stride, 16-wide LDS rows

// ---- WMMA: V_WMMA_F32_16X16X4_F32 (fp32 matrix core, matches ref numerics)
__device__ __forceinline__ v8f wmma_f32(v2f a, v2f b, v8f c) {
  return __builtin_amdgcn_wmma_f32_16x16x4_f32(false, a, false, b, (short)0, c,
                                               false, false);
}

// A-operand fragment from row-major [M=16][K] LDS matrix (also B=Xᵀ loads).
__device__ __forceinline__ v2f fragA(const float* m, int stride, int k0) {
  int lane = threadIdx.x & 31;
  const float* p = m + (lane & 15) * stride + k0 + ((lane >> 4) << 1);
  v2f f; f[0] = p[0]; f[1] = p[1];
  return f;
}
// B-operand fragment from row-major [K][N] LDS matrix; lane column = nbase+lane%16
__device__ __forceinline__ v2f fragB_kn(const float* m, int stride, int k0, int nbase) {
  int lane = threadIdx.x & 31;
  int kk = k0 + ((lane >> 4) << 1);
  int n  = nbase + (lane & 15);
  v2f f; f[0] = m[kk * stride + n]; f[1] = m[(kk + 1) * stride + n];
  return f;
}

// =====================================================================
// Pass 1: per-segment local transfer function.
//   A_seg[i]   = prod of w over segment (w'_0 := 1 for global t==0)
//   S_loc[i,j] = sum_t (prod_{tau>t} w[i]) k_t[i] v_t[j]
// grid (SEG, H), 128 thr. WMMA for the K^·V rank-16 updates.
// =====================================================================
__global__ __launch_bounds__(128)
void wkv_pass1(const float* __restrict__ wg, const float* __restrict__ kg,
               const float* __restrict__ vg, float* __restrict__ Aseg,
               float* __restrict__ Sloc) {
  __shared__ float S[dh * ST];
  __shared__ float Vv[CH * ST];
  __shared__ float Kh[dh * ST16];
  __shared__ float aLast[dh];
  __shared__ float aAcc[dh];

  const int seg = blockIdx.x, h = blockIdx.y;
  const int tid = threadIdx.x;
  const int lane = tid & 31, wid = tid >> 5;
  const int hi = lane >> 4, ln16 = lane & 15;

  for (int x = tid; x < dh * dh; x += 128) S[(x >> 6) * ST + (x & 63)] = 0.f;
  if (tid < dh) aAcc[tid] = 1.f;

  for (int c = 0; c < NCHS; ++c) {
    const int t0 = seg * TSEG + c * CH;
    __syncthreads();
    if (tid < dh) {               // stage: one divide per (t,i) via inverse track
      int i = tid;
      float a = 1.f, ai = 1.f;
      float kr[CH], air[CH];
#pragma unroll
      for (int t = 0; t < CH; ++t) {
        int off  = (t0 + t) * D_ + h * dh + i;
        float wv = wg[off];
        if (seg == 0 && c == 0 && t == 0) wv = 1.f;  // fold w0 into S_init
        float kv = kg[off];
        Vv[t * ST + i] = vg[off];
        float rw = 1.f / wv;
        a *= wv; ai *= rw;
        kr[t] = kv; air[t] = ai;
      }
      aLast[i] = a;
      aAcc[i] *= a;
#pragma unroll
      for (int t = 0; t < CH; ++t) Kh[i * ST16 + t] = kr[t] * (a * air[t]);
    }
    __syncthreads();

    const int nb = wid * 16;      // each wave owns 16 columns: no cross-wave hazard
#pragma unroll
    for (int it = 0; it < 4; ++it) {
      v8f acc = {};
#pragma unroll
      for (int k0 = 0; k0 < CH; k0 += 4)
        acc = wmma_f32(fragA(Kh + it * 16 * ST16, ST16, k0),
                       fragB_kn(Vv, ST, k0, nb), acc);
#pragma unroll
      for (int v = 0; v < 8; ++v) {
        int i   = it * 16 + v + 8 * hi;
        int idx = i * ST + nb + ln16;
        S[idx]  = aLast[i] * S[idx] + acc[v];
      }
    }
  }
  __syncthreads();
  if (tid < dh) Aseg[(h * SEG + seg) * dh + tid] = aAcc[tid];
  for (int x = tid; x < dh * dh; x += 128) {
    int i = x >> 6, j = x & 63;
    Sloc[((size_t)(h * SEG + seg) * dh + i) * dh + j] = S[i * ST + j];
  }
}

// =====================================================================
// Inter-segment scan: S_start[seg+1] = diag(A_seg) S_start[seg] + S_loc[seg].
// Also emits the final-state output directly. grid (H), VALU-only, tiny.
// =====================================================================
__global__ __launch_bounds__(128)
void state_scan(const float* __restrict__ wg, const float* __restrict__ isg,
                const float* __restrict__ Aseg, const float* __restrict__ Sloc,
                float* __restrict__ Sstart, float* __restrict__ state_out) {
  const int h = blockIdx.x, tid = threadIdx.x;
  for (int x = tid; x < dh * dh; x += 128) {
    int i = x >> 6, j = x & 63;
    float s = wg[h * dh + j] * isg[h * dh + j];   // ref's broadcast-row init quirk
    for (int sg = 0; sg < SEG; ++sg) {
      size_t base = ((size_t)(h * SEG + sg) * dh + i) * dh + j;
      Sstart[base] = s;
      s = Aseg[(h * SEG + sg) * dh + i] * s + Sloc[base];
    }
    state_out[i * D_ + h * dh + j] = s;
  }
}

// =====================================================================
// Pass 2: outputs. Chunked WMMA WKV with known segment start state,
// fused LN + SiLU gate. grid (SEG, H), 128 thr (4 waves, 16 cols each).
// =====================================================================
__global__ __launch_bounds__(128)
void wkv_pass2(const float* __restrict__ rg, const float* __restrict__ wg,
               const float* __restrict__ kg, const float* __restrict__ vg,
               const float* __restrict__ gg, const float* __restrict__ ug,
               const float* __restrict__ isg, const float* __restrict__ lnw,
               const float* __restrict__ lnb, const float* __restrict__ Sstart,
               float* __restrict__ og) {
  __shared__ float S[dh * ST];
  __shared__ float Rt[CH * ST];
  __shared__ float Kt[CH * ST];
  __shared__ float Ku[CH * ST];
  __shared__ float Vv[CH * ST];
  __shared__ float Kh[dh * ST16];
  __shared__ float CmW[4][CH * ST16];
  __shared__ float ybuf[CH * ST];
  __shared__ float aLast[dh];
  __shared__ float uvec[dh];
  __shared__ float corr0[dh];
  __shared__ float lnw_s[dh], lnb_s[dh];
  __shared__ float mu_s[CH], rs_s[CH];
  __shared__ float r0sum_s;

  const int seg = blockIdx.x, h = blockIdx.y;
  const int tid = threadIdx.x;
  const int lane = tid & 31, wid = tid >> 5;
  const int hi = lane >> 4, ln16 = lane & 15;

  for (int x = tid; x < dh; x += 128) {
    uvec[x]  = ug[h * dh + x];
    corr0[x] = isg[h * dh + x] * (1.f - wg[h * dh + x]);
    lnw_s[x] = lnw[x];
    lnb_s[x] = lnb[x];
  }
  for (int x = tid; x < dh * dh; x += 128) {
    int i = x >> 6, j = x & 63;
    S[i * ST + j] = Sstart[((size_t)(h * SEG + seg) * dh + i) * dh + j];
  }
  if (tid == 0 && seg == 0) {
    float s = 0.f;
    for (int i2 = 0; i2 < dh; ++i2) s += rg[h * dh + i2];
    r0sum_s = s;
  }
  __syncthreads();

  for (int c = 0; c < NCHS; ++c) {
    const int t0 = seg * TSEG + c * CH;

    if (tid < dh) {               // stage (one divide per (t,i))
      int i = tid;
      float a = 1.f, ai = 1.f;
      float kr[CH], air[CH];
#pragma unroll
      for (int t = 0; t < CH; ++t) {
        int off  = (t0 + t) * D_ + h * dh + i;
        float wv = wg[off];
        if (seg == 0 && c == 0 && t == 0) wv = 1.f;
        float rv = rg[off], kv = kg[off];
        float aip = ai;
        Rt[t * ST + i] = rv * a;               // r * A_{t-1}
        float rw = 1.f / wv;
        a *= wv; ai *= rw;
        Kt[t * ST + i] = kv * ai;              // k / A_t
        Ku[t * ST + i] = kv * uvec[i] * aip;   // u*k / A_{t-1}
        Vv[t * ST + i] = vg[off];
        kr[t] = kv; air[t] = ai;
      }
      aLast[i] = a;
#pragma unroll
      for (int t = 0; t < CH; ++t) Kh[i * ST16 + t] = kr[t] * (a * air[t]);
    }
    __syncthreads();

    // C = R~ K~ᵀ  and diag = R~ Kuᵀ ; causal-mask merge (per-wave copy)
    v8f accC = {}; v8f accD = {};
#pragma unroll
    for (int k0 = 0; k0 < dh; k0 += 4) {
      v2f arf = fragA(Rt, ST, k0);
      accC = wmma_f32(arf, fragA(Kt, ST, k0), accC);
      accD = wmma_f32(arf, fragA(Ku, ST, k0), accD);
    }
    {
      float* cm = CmW[wid];
#pragma unroll
      for (int v = 0; v < 8; ++v) {
        int M = v + 8 * hi, N = ln16;
        cm[M * ST16 + N] = (N < M) ? accC[v] : ((N == M) ? accD[v] : 0.f);
      }
    }
    asm volatile("" ::: "memory");  // same-wave LDS store->load ordering

    // Y = R~ S0 + Cm V   (this wave's 16 columns)
    const int nb = wid * 16;
    v8f accY = {};
#pragma unroll
    for (int k0 = 0; k0 < dh; k0 += 4)
      accY = wmma_f32(fragA(Rt, ST, k0), fragB_kn(S, ST, k0, nb), accY);
#pragma unroll
    for (int k0 = 0; k0 < CH; k0 += 4)
      accY = wmma_f32(fragA(CmW[wid], ST16, k0), fragB_kn(Vv, ST, k0, nb), accY);
    if (seg == 0 && c == 0 && hi == 0)   // broadcast-init-state exact fix, row 0
      accY[0] += r0sum_s * corr0[nb + ln16];
#pragma unroll
    for (int v = 0; v < 8; ++v) ybuf[(v + 8 * hi) * ST + nb + ln16] = accY[v];

    // state-update accumulators K^ V
    v8f accS[4];
#pragma unroll
    for (int it = 0; it < 4; ++it) {
      v8f a = {};
#pragma unroll
      for (int k0 = 0; k0 < CH; k0 += 4)
        a = wmma_f32(fragA(Kh + it * 16 * ST16, ST16, k0),
                     fragB_kn(Vv, ST, k0, nb), a);
      accS[it] = a;
    }
    __syncthreads();   // all waves done reading S; ybuf complete

    // S = diag(A15) S + K^V
#pragma unroll
    for (int it = 0; it < 4; ++it) {
#pragma unroll
      for (int v = 0; v < 8; ++v) {
        int i   = it * 16 + v + 8 * hi;
        int idx = i * ST + nb + ln16;
        S[idx]  = aLast[i] * S[idx] + accS[it][v];
      }
    }
    if (tid < CH) {    // LN stats
      float s = 0.f, sq = 0.f;
      for (int j = 0; j < dh; ++j) {
        float y = ybuf[tid * ST + j]; s += y; sq += y * y;
      }
      float m  = s * (1.f / dh);
      float vv = sq * (1.f / dh) - m * m;
      mu_s[tid] = m;
      rs_s[tid] = rsqrtf(vv + 1e-5f);
    }
    __syncthreads();

    {  // LN + SiLU(g), write o in (T, D) layout
      int t  = tid >> 3;
      int jb = (tid & 7) * 8;
      float m = mu_s[t], rs = rs_s[t];
#pragma unroll
      for (int q = 0; q < 8; ++q) {
        int j    = jb + q;
        float y  = ybuf[t * ST + j];
        float yn = (y - m) * rs * lnw_s[j] + lnb_s[j];
        int off  = (t0 + t) * D_ + h * dh + j;
        float gv = gg[off];
        og[off]  = yn * (gv / (1.f + __expf(-gv)));
      }
    }
    __syncthreads();
  }
}

// =====================================================================
// out = o @ Woᵀ  (4096x1024 @ 1024x1024), fp32 WMMA GEMM, 64x64 tiles.
// =====================================================================
__global__ __launch_bounds__(128)
void oproj_kernel(const float* __restrict__ o, const float* __restrict__ Wo,
                  float* __restrict__ out) {
  __shared__ float Ao[64 * ST];
  __shared__ float Bo[64 * ST];
  const int tid = threadIdx.x;
  const int lane = tid & 31, wid = tid >> 5;
  const int hi = lane >> 4, ln16 = lane & 15;
  const int nbase = blockIdx.x * 64;
  const int mbase = blockIdx.y * 64;

  v8f acc[4] = {};

  for (int ks = 0; ks < D_; ks += 64) {
    __syncthreads();
    {
      int row = tid >> 1;
      int cb  = (tid & 1) * 32;
      const float4* pa = (const float4*)(o  + (size_t)(mbase + row) * D_ + ks + cb);
      const float4* pb = (const float4*)(Wo + (size_t)(nbase + row) * D_ + ks + cb);
#pragma unroll
      for (int q = 0; q < 8; ++q) {
        float4 a4 = pa[q]; float4 b4 = pb[q];
        int cc = cb + q * 4;
        Ao[row * ST + cc + 0] = a4.x; Ao[row * ST + cc + 1] = a4.y;
        Ao[row * ST + cc + 2] = a4.z; Ao[row * ST + cc + 3] = a4.w;
        Bo[row * ST + cc + 0] = b4.x; Bo[row * ST + cc + 1] = b4.y;
        Bo[row * ST + cc + 2] = b4.z; Bo[row * ST + cc + 3] = b4.w;
      }
    }
    __syncthreads();
#pragma unroll
    for (int k0 = 0; k0 < 64; k0 += 4) {
      v2f b = fragA(Bo + wid * 16 * ST, ST, k0);  // B = Woᵀ => A-style row load
#pragma unroll
      for (int mt = 0; mt < 4; ++mt)
        acc[mt] = wmma_f32(fragA(Ao + mt * 16 * ST, ST, k0), b, acc[mt]);
    }
  }
#pragma unroll
  for (int mt = 0; mt < 4; ++mt) {
#pragma unroll
    for (int v = 0; v < 8; ++v) {
      int row = mbase + mt * 16 + v + 8 * hi;
      int col = nbase + wid * 16 + ln16;
      out[(size_t)row * D_ + col] = acc[mt][v];
    }
  }
}

extern "C" void kernel_launch(void* const* d_in, const int* in_sizes, int n_in,
                              void* d_out, int out_size, void* d_ws, size_t ws_size,
                              hipStream_t stream) {
  const float* r   = (const float*)d_in[0];
  const float* w   = (const float*)d_in[1];
  const float* k   = (const float*)d_in[2];
  const float* v   = (const float*)d_in[3];
  const float* g   = (const float*)d_in[4];
  const float* u   = (const float*)d_in[5];
  const float* is  = (const float*)d_in[6];
  const float* lnw = (const float*)d_in[7];
  const float* lnb = (const float*)d_in[8];
  const float* Wo  = (const float*)d_in[9];

  float* out   = (float*)d_out;
  float* state = out + (size_t)T_ * D_;          // outputs: out, then state

  float* o      = (float*)d_ws;                  // T*D
  float* Aseg   = o + (size_t)T_ * D_;           // H*SEG*dh
  float* Sloc   = Aseg + (size_t)H_ * SEG * dh;  // H*SEG*dh*dh
  float* Sstart = Sloc + (size_t)H_ * SEG * dh * dh;

  dim3 gs(SEG, H_);
  wkv_pass1<<<gs, 128, 0, stream>>>(w, k, v, Aseg, Sloc);
  state_scan<<<H_, 128, 0, stream>>>(w, is, Aseg, Sloc, Sstart, state);
  wkv_pass2<<<gs, 128, 0, stream>>>(r, w, k, v, g, u, is, lnw, lnb, Sstart, o);

  dim3 g2(D_ / 64, T_ / 64);
  oproj_kernel<<<g2, 128, 0, stream>>>(o, Wo, out);
}